// Inv1x1MM_SVD_1185410974034
// MI455X (gfx1250) — compile-verified
//
#include <hip/hip_runtime.h>
#include <math.h>

typedef __attribute__((ext_vector_type(2))) float v2f;
typedef __attribute__((ext_vector_type(8))) float v8f;

#define WAVES_PER_BLOCK 8
// per-wave LDS: 16 Q1 columns at stride 18 floats (bank-conflict free, 8B
// aligned pairs) = 288, then 16 floats of y' (8B aligned), total 304.
#define LDS_STRIDE 304
#define YOFF 288

// Broadcast lane k of each 16-lane row to every lane of that row.
// Pure VALU (v_permlane16_b32), no LDS pipe, no dscnt wait -- lanes 0-15
// read lane k, lanes 16-31 read lane 16+k, which is exactly the two-matrix
// (Q0 | Q1) half-wave layout used below.
__device__ __forceinline__ float bcast16(float v, int k) {
  const int sel = k * 0x11111111;  // all 16 select-nibbles = k
  return __int_as_float(__builtin_amdgcn_permlane16(
      __float_as_int(v), __float_as_int(v), sel, sel, false, false));
}

__global__ __launch_bounds__(256) void inv1x1_svd_kernel(
    const float* __restrict__ data,
    const float* __restrict__ paras,
    float* __restrict__ out,
    int total)
{
  __shared__ float lds[WAVES_PER_BLOCK * LDS_STRIDE];

  const int wave = threadIdx.x >> 5;
  const int lane = threadIdx.x & 31;
  const int j    = lane & 15;   // column index within a matrix
  const int half = lane >> 4;   // 0 -> U/Q0, 1 -> V/Q1

  int pos = blockIdx.x * WAVES_PER_BLOCK + wave;
  if (pos >= total) pos = total - 1;  // wave-uniform clamp (no divergent exit)

  const float* p = paras + (size_t)pos * 528;

  // Column j of U (half==0) or V (half==1); u_v is row-major CxC.  Kept as
  // 8 x float2 so the MGS math maps onto packed V_PK_FMA_F32 (2 fp32 per
  // VALU issue).
  v2f a2[8];
  const float* colp = p + 16 + half * 256 + j;
#pragma unroll
  for (int r = 0; r < 8; ++r) {
    a2[r].x = colp[(2 * r) * 16];
    a2[r].y = colp[(2 * r + 1) * 16];
  }

  // Modified Gram-Schmidt, column-per-lane, branchless, no explicit pivot
  // normalization pass: project against the unnormalized broadcast column u
  // with coefficient (u.a)/||u||^2; the pivot lane folds its own
  // normalization into the same FMA pass (u == a there, c = 1 - 1/||u||).
  // r_kk = ||a_k|| > 0, so this Q equals the sign-corrected Q of the
  // reference (QR with positive-diagonal R is unique).
#pragma unroll
  for (int k = 0; k < 16; ++k) {
    v2f u[8];
#pragma unroll
    for (int i = 0; i < 8; ++i) {
      u[i].x = bcast16(a2[i].x, k);
      u[i].y = bcast16(a2[i].y, k);
    }
    v2f n2v = {0.f, 0.f};
#pragma unroll
    for (int i = 0; i < 8; ++i) n2v = __builtin_elementwise_fma(u[i], u[i], n2v);
    const float n2   = n2v.x + n2v.y;
    const float rinv = __builtin_amdgcn_rsqf(n2);   // v_rsq_f32, ~1 ulp
    v2f dv2 = {0.f, 0.f};
#pragma unroll
    for (int i = 0; i < 8; ++i) dv2 = __builtin_elementwise_fma(u[i], a2[i], dv2);
    const float dot = dv2.x + dv2.y;
    // j>k: c = (u.a)/||u||^2   j==k: c = 1-1/||u||  -> a := a/||u||   j<k: 0
    const float c = (j > k) ? dot * (rinv * rinv)
                            : ((j == k) ? (1.0f - rinv) : 0.0f);
    const v2f cn = {-c, -c};
#pragma unroll
    for (int i = 0; i < 8; ++i) a2[i] = __builtin_elementwise_fma(cn, u[i], a2[i]);
  }

  float* wl = &lds[wave * LDS_STRIDE];
  if (half) {
    // Stash Q1 column j at stride 18: 8B-aligned pairs (b64-mergeable),
    // banks (18*j) % 64 distinct across lanes -> conflict-free.
#pragma unroll
    for (int r = 0; r < 8; ++r) {
      wl[j * 18 + 2 * r]     = a2[r].x;
      wl[j * 18 + 2 * r + 1] = a2[r].y;
    }
  } else {
    // y'_j = exp(ls_j) * (data . Q0[:,j])  -- folds diag(s) into the vector.
    const float* dv = data + (size_t)pos * 16;
    v2f yv = {0.f, 0.f};
#pragma unroll
    for (int i = 0; i < 8; ++i) {
      v2f d;
      d.x = dv[2 * i];
      d.y = dv[2 * i + 1];
      yv = __builtin_elementwise_fma(d, a2[i], yv);
    }
    wl[YOFF + j] = (yv.x + yv.y) * expf(p[j]);
  }
  __syncthreads();

  // res = y' @ Q1^T as a 16x16x16 fp32 matmul:
  // A (16x4 per step) has every row equal to the y' K-slice -> every row of D
  // equals res; B (4x16 per step) = Q1^T rows pulled from the staged columns.
  // ISA layouts: A: M = lane&15, K = vgpr + 2*(lane>>4); B: N = lane&15, same K.
  v8f acc = {};
#pragma unroll
  for (int kb = 0; kb < 4; ++kb) {
    const int k0 = kb * 4 + half * 2;
    v2f va, vb;
    va.x = wl[YOFF + k0];           // y'[k0]   (contiguous, 8B aligned pair)
    va.y = wl[YOFF + k0 + 1];       // y'[k0+1]
    vb.x = wl[k0 * 18 + j];         // Q1^T[k0,   n=j] = Q1[j, k0]
    vb.y = wl[(k0 + 1) * 18 + j];   // Q1^T[k0+1, n=j]
    acc = __builtin_amdgcn_wmma_f32_16x16x4_f32(false, va, false, vb,
                                                (short)0, acc, false, false);
  }

  // D[M=0, N=n] sits in VGPR0 of lanes 0-15.
  if (!half) out[(size_t)pos * 16 + j] = acc[0];
}

extern "C" void kernel_launch(void* const* d_in, const int* in_sizes, int n_in,
                              void* d_out, int out_size, void* d_ws, size_t ws_size,
                              hipStream_t stream) {
  (void)n_in; (void)out_size; (void)d_ws; (void)ws_size;
  const float* data  = (const float*)d_in[0];   // (B,T,C) f32
  const float* paras = (const float*)d_in[1];   // (B,T,528) f32
  float* out = (float*)d_out;                   // (B,T,C) f32
  const int total  = in_sizes[0] / 16;          // B*T = 65536 positions
  const int blocks = (total + WAVES_PER_BLOCK - 1) / WAVES_PER_BLOCK;
  inv1x1_svd_kernel<<<blocks, 256, 0, stream>>>(data, paras, out, total);
}